// DCTExtractor_88794153877655
// MI455X (gfx1250) — compile-verified
//
#include <hip/hip_runtime.h>
#include <hip/hip_bf16.h>

// ---------------------------------------------------------------------------
// DCT LSB extractor for MI455X (gfx1250).
//
// Roofline: 96MB read + 50MB write @ 23.3 TB/s ~= 6.3us -> memory bound.
// Strategy:
//   kernel 1: vectorized zero-fill of d_out (48MB, float4 stores).
//   kernel 2: per-wave 16x16 tile (= 2x2 of 8x8 DCT blocks).
//     T  = D2 @ X     via 4x chained V_WMMA_F32_16X16X4_F32   (f32 precision)
//     Y' = D2 @ T^T   via 4x chained V_WMMA_F32_16X16X4_F32   (Y' = (D X D^T)^T)
//     T^T formed by an LDS round-trip (stride-18 rows -> bank-conflict free).
//     The 16 wanted coefficients are routed with ds_bpermute_b32 and lanes
//     0..15 store |rint(c)| mod 2 straight into the owning output segment.
// D2 = blockdiag(D, D) computes all four 8x8 block DCTs of the tile at once.
// ---------------------------------------------------------------------------

typedef float v2f __attribute__((ext_vector_type(2)));
typedef float v8f __attribute__((ext_vector_type(8)));

constexpr int Bdim = 8, Cdim = 3, Hdim = 1024, Wdim = 1024;
constexpr int NBH = Hdim / 8, NBW = Wdim / 8;          // 128 x 128 blocks
constexpr int TH = Hdim / 16, TW = Wdim / 16;          // 64 x 64 tiles
constexpr int NT = Bdim * Cdim * TH * TW;              // 98304 tiles
constexpr int NUM_BITS = Bdim * Cdim * NBH * NBW * 4;  // 1572864
constexpr int OUT_ELEMS = Bdim * NUM_BITS;             // 12582912

// Orthonormal 8x8 DCT-II matrix, row-major D[k][m].
__device__ __constant__ float c_D[64] = {
  0.3535533905932738f,  0.3535533905932738f,  0.3535533905932738f,  0.3535533905932738f,
  0.3535533905932738f,  0.3535533905932738f,  0.3535533905932738f,  0.3535533905932738f,
  0.4903926402016152f,  0.4157348061512726f,  0.2777851165098011f,  0.0975451610080641f,
 -0.0975451610080641f, -0.2777851165098011f, -0.4157348061512726f, -0.4903926402016152f,
  0.4619397662556434f,  0.1913417161825449f, -0.1913417161825449f, -0.4619397662556434f,
 -0.4619397662556434f, -0.1913417161825449f,  0.1913417161825449f,  0.4619397662556434f,
  0.4157348061512726f, -0.0975451610080641f, -0.4903926402016152f, -0.2777851165098011f,
  0.2777851165098011f,  0.4903926402016152f,  0.0975451610080641f, -0.4157348061512726f,
  0.3535533905932738f, -0.3535533905932738f, -0.3535533905932738f,  0.3535533905932738f,
  0.3535533905932738f, -0.3535533905932738f, -0.3535533905932738f,  0.3535533905932738f,
  0.2777851165098011f, -0.4903926402016152f,  0.0975451610080641f,  0.4157348061512726f,
 -0.4157348061512726f, -0.0975451610080641f,  0.4903926402016152f, -0.2777851165098011f,
  0.1913417161825449f, -0.4619397662556434f,  0.4619397662556434f, -0.1913417161825449f,
 -0.1913417161825449f,  0.4619397662556434f, -0.4619397662556434f,  0.1913417161825449f,
  0.0975451610080641f, -0.2777851165098011f,  0.4157348061512726f, -0.4903926402016152f,
  0.4903926402016152f, -0.4157348061512726f,  0.2777851165098011f, -0.0975451610080641f
};

// -------------------------------- zero fill --------------------------------
__global__ void dct_zero_kernel(float4* __restrict__ out, int n4) {
  int i = blockIdx.x * blockDim.x + threadIdx.x;
  if (i < n4) out[i] = make_float4(0.f, 0.f, 0.f, 0.f);
}

// ----------------------------- DCT + extract -------------------------------
__global__ void __launch_bounds__(256)
dct_extract_kernel(const float* __restrict__ x, float* __restrict__ out) {
  // per-wave LDS tile for the T -> T^T round trip; stride 18 floats:
  // even (float2-aligned) and 18*m % 64 distinct over m=0..15 -> no conflicts.
  __shared__ float lds_T[8][16 * 18];

  const int lane = threadIdx.x & 31;
  const int wib  = threadIdx.x >> 5;
  const int gw   = (blockIdx.x * blockDim.x + threadIdx.x) >> 5;
  const int nwv  = (gridDim.x * blockDim.x) >> 5;

  const int m    = lane & 15;   // row (A) / column (B,C/D) index
  const int half = lane >> 4;   // K-half selector for A/B operands
  const int koff = 2 * half;

  // A operand = D2 = blockdiag(D,D), split into 4 chunks of K=4.
  // ISA layout (32-bit A 16x4): lane m holds K = {chunk*4+koff, chunk*4+koff+1}.
  v2f a[4];
#pragma unroll
  for (int k = 0; k < 4; ++k) {
#pragma unroll
    for (int p = 0; p < 2; ++p) {
      int j = 4 * k + koff + p;                 // column of D2, 0..15
      a[k][p] = ((m >> 3) == (j >> 3)) ? c_D[(m & 7) * 8 + (j & 7)] : 0.0f;
    }
  }

  // Extraction mapping: lane l<16 handles (r,s,p); coefficient (u_p,v_p)
  // of sub-block (r,s) lives in Y'^(D-layout) at VGPR v_p, lane 16s+8r+u_p.
  const int p_ = lane & 3;
  const int s_ = (lane >> 2) & 1;
  const int r_ = (lane >> 3) & 1;
  const int u_ = (p_ == 0) ? 1 : (p_ == 3 ? 3 : 2);   // {1,2,2,3}
  const int v_ = (p_ & 1) ? 1 : 2;                    // {2,1,2,1}
  const int src_lane = 16 * s_ + 8 * r_ + u_;

  float* T = lds_T[wib];

  for (int t = gw; t < NT; t += nwv) {            // uniform per-wave loop
    const int tw = t & (TW - 1);
    const int th = (t >> 6) & (TH - 1);
    const int c  = (t >> 12) % Cdim;
    const int b  = t / (TW * TH * Cdim);
    const float* base =
        x + (((size_t)(b * Cdim + c)) * Hdim + (size_t)th * 16) * Wdim + tw * 16;

    // B operand = X tile rows, ISA B 4x16 layout:
    // chunk k, VGPR p, lane l -> X[4k + p + 2*(l>>4)][l&15]  (coalesced rows)
    v2f bx[4];
#pragma unroll
    for (int k = 0; k < 4; ++k)
#pragma unroll
      for (int p = 0; p < 2; ++p)
        bx[k][p] = base[(size_t)(4 * k + p + koff) * Wdim + m];

    // T = D2 @ X  (full f32 precision)
    v8f acc = {0.f, 0.f, 0.f, 0.f, 0.f, 0.f, 0.f, 0.f};
#pragma unroll
    for (int k = 0; k < 4; ++k)
      acc = __builtin_amdgcn_wmma_f32_16x16x4_f32(
          false, a[k], false, bx[k], (short)0, acc, false, false);

    // C/D layout: VGPR w, lane l -> T[w + 8*(l>>4)][l&15]; store row-major.
#pragma unroll
    for (int w = 0; w < 8; ++w)
      T[(w + 8 * half) * 18 + m] = acc[w];
    // same-wave LDS ops are in-order (DScnt ordering) -> no barrier needed.

    // B operand = T^T: chunk k, VGPR p, lane l -> T[l&15][4k + p + koff]
    v2f bt[4];
#pragma unroll
    for (int k = 0; k < 4; ++k)
#pragma unroll
      for (int p = 0; p < 2; ++p)
        bt[k][p] = T[m * 18 + 4 * k + p + koff];

    // Y' = D2 @ T^T = (D2 X D2^T)^T
    v8f acc2 = {0.f, 0.f, 0.f, 0.f, 0.f, 0.f, 0.f, 0.f};
#pragma unroll
    for (int k = 0; k < 4; ++k)
      acc2 = __builtin_amdgcn_wmma_f32_16x16x4_f32(
          false, a[k], false, bt[k], (short)0, acc2, false, false);

    // Route the 16 wanted coefficients to lanes 0..15 (all lanes active).
    int g1 = __builtin_amdgcn_ds_bpermute(src_lane * 4, __float_as_int(acc2[1]));
    int g2 = __builtin_amdgcn_ds_bpermute(src_lane * 4, __float_as_int(acc2[2]));
    float coeff = __int_as_float((v_ == 1) ? g1 : g2);
    float av  = fabsf(rintf(coeff));            // rint = round-half-even = jnp.round
    float bit = av - 2.0f * floorf(av * 0.5f);  // parity

    if (lane < 16) {
      int nh = 2 * th + r_;
      int nw = 2 * tw + s_;
      size_t i = (size_t)((((b * Cdim + c) * NBH + nh) * NBW + nw) * 4 + p_);
      out[(size_t)b * NUM_BITS + i] = bit;      // only the owning row is nonzero
    }
  }
}

// ------------------------------- launcher ----------------------------------
extern "C" void kernel_launch(void* const* d_in, const int* in_sizes, int n_in,
                              void* d_out, int out_size, void* d_ws, size_t ws_size,
                              hipStream_t stream) {
  (void)in_sizes; (void)n_in; (void)d_ws; (void)ws_size; (void)out_size;
  const float* stego = (const float*)d_in[0];   // [8,3,1024,1024] f32
  float* out = (float*)d_out;                   // [8, NUM_BITS] f32

  // 1) zero the whole output (rows only own their segment; rest stays 0)
  const int n4 = OUT_ELEMS / 4;
  dct_zero_kernel<<<(n4 + 255) / 256, 256, 0, stream>>>((float4*)out, n4);

  // 2) DCT + parity extraction; 8 waves/block, grid-stride over 98304 tiles
  dct_extract_kernel<<<1536, 256, 0, stream>>>(stego, out);
}